// slater_jastrow_37898791420048
// MI455X (gfx1250) — compile-verified
//
#include <hip/hip_runtime.h>
#include <math.h>

// CDNA5 / gfx1250: wave32, WMMA f32 16x16x4 for LU trailing updates.

typedef __attribute__((ext_vector_type(2))) float v2f;
typedef __attribute__((ext_vector_type(8))) float v8f;

#define N_ORB 128
#define NF    64
#define LDA   65          // LDS row stride (floats): conflict-free rows & columns
#define NPAIR 2016        // 64*63/2 occupied pairs
#define PI_F  3.14159265358979323846f

__global__ __launch_bounds__(256) void sj_logpsi_kernel(
    const float* __restrict__ nocc,
    const float* __restrict__ kre,
    const float* __restrict__ kim,
    float* __restrict__ out)
{
    __shared__ float Are[NF][LDA];
    __shared__ float Aim[NF][LDA];
    __shared__ float occ[N_ORB];
    __shared__ int   Ridx[NF];
    __shared__ float pv[64];
    __shared__ int   pidx[64];
    __shared__ float redr[256], redi[256];
    __shared__ float s_ldre, s_ldim;
    __shared__ int   s_nswap;

    const int b   = blockIdx.x;
    const int tid = threadIdx.x;

    if (tid == 0) { s_ldre = 0.f; s_ldim = 0.f; s_nswap = 0; }
    if (tid < N_ORB) occ[tid] = nocc[b * N_ORB + tid];
    __syncthreads();

    // ---- occupied indices, ascending (stable argsort of -n, first NF) ----
    if (tid < N_ORB) {
        if (occ[tid] > 0.5f) {
            int rank = 0;
            for (int j = 0; j < tid; ++j) rank += (occ[j] > 0.5f) ? 1 : 0;
            Ridx[rank] = tid;
        }
    }
    __syncthreads();

    // ---- gather A = kernel[R] (complex) into LDS ----
    for (int e = tid; e < NF * NF; e += 256) {
        int r = e >> 6, c = e & 63;
        int row = Ridx[r];
        Are[r][c] = kre[row * NF + c];
        Aim[r][c] = kim[row * NF + c];
    }

    // ---- Jastrow: sum of W[R[a],R[b]] over pairs a>b; W flat index i(i-1)/2+j ----
    float jr = 0.f, ji = 0.f;
    for (int p = tid; p < NPAIR; p += 256) {
        int a = (int)((1.0f + sqrtf(1.0f + 8.0f * (float)p)) * 0.5f);
        while (a * (a - 1) / 2 > p) --a;
        while ((a + 1) * a / 2 <= p) ++a;
        int bb = p - a * (a - 1) / 2;
        int i = Ridx[a], jx = Ridx[bb];
        int q = i * (i - 1) / 2 + jx;          // max 8127 < 8192
        jr += kre[q]; ji += kim[q];
    }
    redr[tid] = jr; redi[tid] = ji;
    __syncthreads();
    for (int s = 128; s > 0; s >>= 1) {        // deterministic tree reduce
        if (tid < s) { redr[tid] += redr[tid + s]; redi[tid] += redi[tid + s]; }
        __syncthreads();
    }

    // ---- blocked complex LU with partial pivoting, panel width 16 ----
    for (int kb = 0; kb < NF; kb += 16) {
        // panel factorization, columns kb..kb+15
        for (int j = kb; j < kb + 16; ++j) {
            // pivot search over rows j..63 (cabs1 norm)
            if (tid < 64) {
                float vr = Are[tid][j], vi = Aim[tid][j];
                pv[tid]   = (tid >= j) ? (fabsf(vr) + fabsf(vi)) : -1.f;
                pidx[tid] = tid;
            }
            __syncthreads();
            if (tid < 32) {
                float v = pv[tid];       int ix  = pidx[tid];
                float v2 = pv[tid + 32]; int ix2 = pidx[tid + 32];
                if (v2 > v) { v = v2; ix = ix2; }
                for (int off = 16; off > 0; off >>= 1) {
                    float ov  = __shfl_down(v, off);
                    int   oix = __shfl_down(ix, off);
                    if (ov > v) { v = ov; ix = oix; }
                }
                if (tid == 0) pidx[0] = ix;
            }
            __syncthreads();
            int piv = pidx[0];
            if (piv != j && tid < 64) {        // swap full rows (re+im)
                float tr = Are[j][tid]; Are[j][tid] = Are[piv][tid]; Are[piv][tid] = tr;
                float ti = Aim[j][tid]; Aim[j][tid] = Aim[piv][tid]; Aim[piv][tid] = ti;
            }
            __syncthreads();
            float ur = Are[j][j], ui = Aim[j][j];
            if (tid == 0) {
                if (piv != j) ++s_nswap;
                s_ldre += 0.5f * logf(ur * ur + ui * ui);
                s_ldim += atan2f(ui, ur);
            }
            {   // scale column below pivot: A[i][j] /= U_jj
                int i = j + 1 + tid;
                if (i < 64) {
                    float inv = 1.f / (ur * ur + ui * ui);
                    float ar = Are[i][j], ai = Aim[i][j];
                    Are[i][j] = (ar * ur + ai * ui) * inv;
                    Aim[i][j] = (ai * ur - ar * ui) * inv;
                }
            }
            __syncthreads();
            // rank-1 update restricted to remaining panel columns
            int ncols = kb + 16 - (j + 1);
            int nrows = 63 - j;
            for (int e = tid; e < ncols * nrows; e += 256) {
                int rr = e / ncols, cc = e - rr * ncols;
                int i = j + 1 + rr, c = j + 1 + cc;
                float lr = Are[i][j], li = Aim[i][j];
                float wr = Are[j][c], wi = Aim[j][c];
                Are[i][c] -= lr * wr - li * wi;
                Aim[i][c] -= lr * wi + li * wr;
            }
            __syncthreads();
        }

        int W = 48 - kb;                       // trailing width
        if (W > 0) {
            // U12 = L11^{-1} * A12 (unit-lower triangular solve), one thread/column
            if (tid < W) {
                int c = kb + 16 + tid;
                for (int ii = 1; ii < 16; ++ii) {
                    float sr = 0.f, si = 0.f;
                    for (int jj = 0; jj < ii; ++jj) {
                        float lr = Are[kb + ii][kb + jj], li = Aim[kb + ii][kb + jj];
                        float wr = Are[kb + jj][c],       wi = Aim[kb + jj][c];
                        sr += lr * wr - li * wi;
                        si += lr * wi + li * wr;
                    }
                    Are[kb + ii][c] -= sr;
                    Aim[kb + ii][c] -= si;
                }
            }
            __syncthreads();

            // A22 -= L21 * U12 : complex GEMM via f32 WMMA 16x16x4 chains.
            // Re(C) += (-Lr)*Ur + Li*Ui ; Im(C) += (-Lr)*Ui + (-Li)*Ur
            int nt   = W >> 4;
            int T    = nt * nt;
            int wave = tid >> 5;
            int lane = tid & 31;
            int m16  = lane & 15;
            int hi   = lane >> 4;              // 0: lanes 0-15, 1: lanes 16-31
            for (int t = wave; t < T; t += 8) {   // wave-uniform -> EXEC all 1s
                int tr = t / nt, tc = t - tr * nt;
                int r0 = kb + 16 + tr * 16;
                int c0 = kb + 16 + tc * 16;
                v8f cre, cim;
                #pragma unroll
                for (int v = 0; v < 8; ++v) {     // C/D layout: VGPR v = row v (+8 for hi)
                    int m = v + hi * 8;
                    cre[v] = Are[r0 + m][c0 + m16];
                    cim[v] = Aim[r0 + m][c0 + m16];
                }
                #pragma unroll
                for (int ks = 0; ks < 4; ++ks) {  // K = 16 in steps of 4
                    int k0 = kb + ks * 4;
                    v2f aRn, aI, aIn, bR, bI;
                    #pragma unroll
                    for (int pq = 0; pq < 2; ++pq) {
                        int kk = pq + 2 * hi;     // A 16x4: vgpr p, lanes>=16 -> K+2
                        float lr = Are[r0 + m16][k0 + kk];
                        float li = Aim[r0 + m16][k0 + kk];
                        aRn[pq] = -lr; aI[pq] = li; aIn[pq] = -li;
                        bR[pq] = Are[k0 + kk][c0 + m16];
                        bI[pq] = Aim[k0 + kk][c0 + m16];
                    }
                    cre = __builtin_amdgcn_wmma_f32_16x16x4_f32(false, aRn, false, bR, (short)0, cre, false, false);
                    cre = __builtin_amdgcn_wmma_f32_16x16x4_f32(false, aI,  false, bI, (short)0, cre, false, false);
                    cim = __builtin_amdgcn_wmma_f32_16x16x4_f32(false, aRn, false, bI, (short)0, cim, false, false);
                    cim = __builtin_amdgcn_wmma_f32_16x16x4_f32(false, aIn, false, bR, (short)0, cim, false, false);
                }
                #pragma unroll
                for (int v = 0; v < 8; ++v) {
                    int m = v + hi * 8;
                    Are[r0 + m][c0 + m16] = cre[v];
                    Aim[r0 + m][c0 + m16] = cim[v];
                }
            }
            __syncthreads();
        }
    }

    // ---- output: log|det| + i*wrap(arg det) + Jastrow ----
    if (tid == 0) {
        float ph = s_ldim + PI_F * (float)s_nswap;
        ph = atan2f(sinf(ph), cosf(ph));       // wrap to (-pi, pi] == log(sign).imag
        out[2 * b]     = s_ldre + redr[0];
        out[2 * b + 1] = ph + redi[0];
    }
}

extern "C" void kernel_launch(void* const* d_in, const int* in_sizes, int n_in,
                              void* d_out, int out_size, void* d_ws, size_t ws_size,
                              hipStream_t stream) {
    const float* n   = (const float*)d_in[0];
    const float* kre = (const float*)d_in[1];
    const float* kim = (const float*)d_in[2];
    float* out = (float*)d_out;
    int B = in_sizes[0] / N_ORB;               // 16384 samples
    sj_logpsi_kernel<<<B, 256, 0, stream>>>(n, kre, kim, out);
}